// GrapsuleNet_60601988546918
// MI455X (gfx1250) — compile-verified
//
#include <hip/hip_runtime.h>
#include <hip/hip_bf16.h>
#include <math.h>

// Shapes (from reference): B=2, N=1024, C=H=O=64, D=2.
// out[b,i,o] = (1/N) * sum_h W1[o,h] * S[b,i,h] + b1[o]
//   S[b,i,h]  = sum_j silu( lx[b,j,h] * (ea0*We[h,0] + ea1*We[h,1]) )
//   lx[b,j,h] = sum_c x[b,j,c]*W0[h,c] + b0[h]

typedef __attribute__((ext_vector_type(2))) float v2f;
typedef __attribute__((ext_vector_type(8))) float v8f;

// ---------------------------------------------------------------------------
// Generic tiny GEMM with V_WMMA_F32_16X16X4_F32:
//   D[m,n] = alpha * sum_{k=0..63} A[m,k] * W[n,k]  + bias[n]
// A: [M x 64] row-major, W: [64 x 64] row-major (n-major, i.e. B[k][n]=W[n*64+k])
// One wave (32 threads) per 16x16 output tile. grid = (M/16, 64/16), block = 32.
// Fragment layouts per CDNA5 ISA 7.12.2:
//   A (16x4 f32): lanes 0-15 hold row M=lane, {v0,v1}={K0,K1}; lanes 16-31 hold
//                 the same rows with {v0,v1}={K2,K3}.
//   B (4x16 f32): mirrored — lane half selects K-pair, lane%16 selects column.
//   C/D (16x16):  VGPR r -> M = r + 8*(lane>=16), N = lane%16.
// ---------------------------------------------------------------------------
__global__ void __launch_bounds__(32)
wmma_gemm_nt64(const float* __restrict__ A,
               const float* __restrict__ W,
               const float* __restrict__ bias,
               float* __restrict__ D,
               float alpha)
{
    const int m0   = blockIdx.x * 16;
    const int n0   = blockIdx.y * 16;
    const int lane = threadIdx.x;          // 0..31, wave32
    const int lr   = lane & 15;
    const int hi   = lane >> 4;            // 0: lanes 0-15, 1: lanes 16-31
    const int kh   = hi * 2;               // K offset for upper lane half

    const float* __restrict__ arow = A + (size_t)(m0 + lr) * 64;
    const float* __restrict__ wrow = W + (size_t)(n0 + lr) * 64;  // column n=n0+lr of B

    v8f c = {};
#pragma unroll
    for (int k = 0; k < 64; k += 4) {
        v2f a, b;
        a.x = arow[k + kh + 0];
        a.y = arow[k + kh + 1];
        b.x = wrow[k + kh + 0];
        b.y = wrow[k + kh + 1];
        // 8 args: (neg_a, A, neg_b, B, c_mod, C, reuse_a, reuse_b)
        c = __builtin_amdgcn_wmma_f32_16x16x4_f32(false, a, false, b,
                                                  (short)0, c, false, false);
    }

    const float bn = bias[n0 + lr];
#pragma unroll
    for (int r = 0; r < 8; ++r) {
        D[(size_t)(m0 + r + hi * 8) * 64 + (n0 + lr)] = alpha * c[r] + bn;
    }
}

// ---------------------------------------------------------------------------
// Dominant kernel: S[b,i,h] = sum_j silu( lx[b,j,h] * (e0*We[h,0]+e1*We[h,1]) )
// One block handles TI=4 consecutive flat rows (b*N+i). 256 threads = 8 waves.
// Edge rows staged in LDS (coalesced float2 loads, 32 KB); each lx[b,j,h]
// value (L2-resident, 512 KB total) is reused across the TI accumulators.
// Thread owns h = tid&63 and j-stride-4 slice; 4-way cross-i ILP hides v_exp.
// ---------------------------------------------------------------------------
#define TI    4
#define NMAX  1024

__global__ void __launch_bounds__(256)
silu_reduce_kernel(const float* __restrict__ edge,  // [B,N,N,2]
                   const float* __restrict__ lx,    // [B,N,64]
                   const float* __restrict__ We,    // [64,2]
                   float* __restrict__ S,           // [B,N,64]
                   int N)
{
    __shared__ float2 eLDS[TI][NMAX];   // 32 KB
    __shared__ float  red[TI][256];     //  4 KB

    const int tid  = threadIdx.x;
    const int base = blockIdx.x * TI;   // flat (b*N + i) base; TI divides N
    const int b    = base / N;

    // Stage TI edge rows: edge[b,i,j,0:2] as float2, flat index (b*N+i)*N + j
    const float2* __restrict__ ep = (const float2*)edge + (size_t)base * N;
    for (int t = tid; t < TI * N; t += 256) {
        eLDS[t / NMAX][t % NMAX] = ep[(size_t)(t / N) * N + (t % N)];
    }

    const int   h   = tid & 63;
    const int   jg  = tid >> 6;                 // 0..3 j-phase
    const float we0 = We[h * 2 + 0];
    const float we1 = We[h * 2 + 1];

    float acc[TI] = {0.f, 0.f, 0.f, 0.f};
    __syncthreads();

    const float* __restrict__ lxb = lx + (size_t)b * N * 64;
#pragma unroll 2
    for (int j = jg; j < N; j += 4) {
        const float lxv = lxb[(size_t)j * 64 + h];   // coalesced per 32-lane wave
#pragma unroll
        for (int ii = 0; ii < TI; ++ii) {
            const float2 e = eLDS[ii][j];            // wave-uniform -> LDS broadcast
            const float  m = lxv * fmaf(e.x, we0, e.y * we1);
            // silu(m) = m * sigmoid(m) = m / (1 + exp(-m))
            acc[ii] += m * __builtin_amdgcn_rcpf(1.0f + __expf(-m));
        }
    }

#pragma unroll
    for (int ii = 0; ii < TI; ++ii) red[ii][tid] = acc[ii];
    __syncthreads();

    if (tid < 64) {
#pragma unroll
        for (int ii = 0; ii < TI; ++ii) {
            const float s = red[ii][tid] + red[ii][tid + 64] +
                            red[ii][tid + 128] + red[ii][tid + 192];
            S[(size_t)(base + ii) * 64 + tid] = s;
        }
    }
}

// ---------------------------------------------------------------------------
extern "C" void kernel_launch(void* const* d_in, const int* in_sizes, int n_in,
                              void* d_out, int out_size, void* d_ws, size_t ws_size,
                              hipStream_t stream)
{
    const float* x    = (const float*)d_in[0];  // [B,N,64]
    const float* edge = (const float*)d_in[1];  // [B,N,N,2]
    const float* W0   = (const float*)d_in[2];  // [64,64]
    const float* b0   = (const float*)d_in[3];  // [64]
    const float* We   = (const float*)d_in[4];  // [64,2]
    const float* W1   = (const float*)d_in[5];  // [64,64]
    const float* b1   = (const float*)d_in[6];  // [64]
    float*       out  = (float*)d_out;          // [B,N,64]

    // Derive shapes: in_sizes[0] = B*N*64, in_sizes[1] = B*N*N*2  => N = 32*ratio
    const int N = (int)(32ll * in_sizes[1] / in_sizes[0]);
    const int M = in_sizes[0] / 64;             // B*N flat rows

    // Workspace: lx [M x 64] then S [M x 64]  (1 MB total for B=2,N=1024)
    float* lx = (float*)d_ws;
    float* S  = lx + (size_t)M * 64;

    // 1) lx = x @ W0^T + b0           (WMMA f32 16x16x4)
    dim3 ggemm(M / 16, 64 / 16);
    wmma_gemm_nt64<<<ggemm, 32, 0, stream>>>(x, W0, b0, lx, 1.0f);

    // 2) S[b,i,h] = sum_j silu(lx * emb)   (dominant kernel; trans/VALU bound)
    silu_reduce_kernel<<<M / TI, 256, 0, stream>>>(edge, lx, We, S, N);

    // 3) out = (1/N) * S @ W1^T + b1  (WMMA f32 16x16x4)
    wmma_gemm_nt64<<<ggemm, 32, 0, stream>>>(S, W1, b1, out, 1.0f / (float)N);
}